// DP_9723805958320
// MI455X (gfx1250) — compile-verified
//
#include <hip/hip_runtime.h>

#define BB    4
#define NAA   1024
#define NPTC  512
#define MAXNC 100
#define MMC   200
#define EMBC  25
#define DDC   400
#define FHC   50
#define SCALEF (1.0f/200.0f)

typedef __attribute__((ext_vector_type(2))) float v2f;
typedef __attribute__((ext_vector_type(8))) float v8f;

__device__ __forceinline__ v8f wmma4(v2f a, v2f b, v8f c) {
  // V_WMMA_F32_16X16X4_F32: D = A(16x4) * B(4x16) + C(16x16)
  return __builtin_amdgcn_wmma_f32_16x16x4_f32(false, a, false, b, (short)0, c, false, false);
}

// ---------------------------------------------------------------- K0: zero out
__global__ void k_zero(float* p, int n) {
  int i = blockIdx.x * blockDim.x + threadIdx.x;
  if (i < n) p[i] = 0.f;
}

// ---------------------------------------------------------------- K1: embedding fwd + xyz
// one wave per atom; 16 neighbor-rows per WMMA tile; weights read from global (L2-resident)
__global__ void __launch_bounds__(128) k_emb_fwd(
    const float* __restrict__ Ri,
    const float* __restrict__ ew0, const float* __restrict__ eb0,
    const float* __restrict__ ew1, const float* __restrict__ eb1,
    const float* __restrict__ ew2, const float* __restrict__ eb2,
    float* __restrict__ xyz_ws)
{
  __shared__ float sT[4][2][16][26];
  __shared__ float sR[4][16][4];
  const int wave = threadIdx.x >> 5, lane = threadIdx.x & 31;
  const int g = lane >> 4, r = lane & 15;
  const int atom = blockIdx.x * 4 + wave;
  const int n = atom & (NAA - 1);
  const int t = (n >= NPTC) ? 1 : 0;
  const float* ribase = Ri + (size_t)atom * MMC * 4;
  const int c1 = 16 + r;

  v8f xc0, xc1;
#pragma unroll
  for (int i = 0; i < 8; i++) { xc0[i] = 0.f; xc1[i] = 0.f; }

  float (*cur)[26] = sT[wave][0];
  float (*nxt)[26] = sT[wave][1];
  float (*Rt)[4]   = sR[wave];

  for (int t1 = 0; t1 < 2; ++t1) {
    const int p = t * 2 + t1;
    const float* w0 = ew0 + p * EMBC;
    const float* b0 = eb0 + p * EMBC;
    const float* W1 = ew1 + p * EMBC * EMBC;
    const float* B1 = eb1 + p * EMBC;
    const float* W2 = ew2 + p * EMBC * EMBC;
    const float* B2 = eb2 + p * EMBC;

    for (int tile = 0; tile < 7; ++tile) {
      const int mrow  = tile * 16 + r;
      const bool valid = (mrow < MAXNC);
      const int mglob = t1 * MAXNC + mrow;

      if (lane < 16) {
        float4 rv = valid ? *(const float4*)(ribase + (size_t)mglob * 4)
                          : make_float4(0.f, 0.f, 0.f, 0.f);
        Rt[r][0] = rv.x; Rt[r][1] = rv.y; Rt[r][2] = rv.z; Rt[r][3] = rv.w;
      }
      const float S = valid ? ribase[(size_t)mglob * 4] : 0.f;

      float h[EMBC];
#pragma unroll
      for (int e = 0; e < EMBC; e++) h[e] = tanhf(S * w0[e] + b0[e]);
      if (lane < 16) {
#pragma unroll
        for (int e = 0; e < EMBC; e++) cur[r][e] = h[e];
      }

      for (int L = 0; L < 2; ++L) {
        const float* W  = L ? W2 : W1;
        const float* Bv = L ? B2 : B1;
        v8f a0, a1;
#pragma unroll
        for (int i = 0; i < 8; i++) { a0[i] = Bv[r]; a1[i] = (c1 < EMBC) ? Bv[c1] : 0.f; }
        for (int kk = 0; kk < 7; ++kk) {
          const int k = kk * 4 + 2 * g;
          v2f av;  av.x  = (k < EMBC) ? h[k] : 0.f;       av.y  = (k + 1 < EMBC) ? h[k + 1] : 0.f;
          v2f bv0; bv0.x = (k < EMBC) ? W[k * EMBC + r] : 0.f;
                   bv0.y = (k + 1 < EMBC) ? W[(k + 1) * EMBC + r] : 0.f;
          v2f bv1; bv1.x = (k < EMBC && c1 < EMBC) ? W[k * EMBC + c1] : 0.f;
                   bv1.y = (k + 1 < EMBC && c1 < EMBC) ? W[(k + 1) * EMBC + c1] : 0.f;
          a0 = wmma4(av, bv0, a0);
          a1 = wmma4(av, bv1, a1);
        }
#pragma unroll
        for (int vr = 0; vr < 8; ++vr) {
          const int row = vr + 8 * g;
          nxt[row][r] = tanhf(a0[vr]) + cur[row][r];
          if (c1 < EMBC) nxt[row][c1] = tanhf(a1[vr]) + cur[row][c1];
        }
        { float (*tmp)[26] = cur; cur = nxt; nxt = tmp; }
#pragma unroll
        for (int e = 0; e < EMBC; e++) h[e] = cur[r][e];
      }

      // xyz accumulation: (G^T)(16x25-part) x R(16x4) via WMMA, R rows of invalid m are zero
      for (int kc = 0; kc < 4; ++kc) {
        const int k = kc * 4 + 2 * g;
        v2f aG0; aG0.x = cur[k][r];  aG0.y = cur[k + 1][r];
        v2f aG1; aG1.x = (c1 < EMBC) ? cur[k][c1] : 0.f;
                 aG1.y = (c1 < EMBC) ? cur[k + 1][c1] : 0.f;
        v2f bR;  bR.x = (r < 4) ? Rt[k][r] : 0.f;  bR.y = (r < 4) ? Rt[k + 1][r] : 0.f;
        xc0 = wmma4(aG0, bR, xc0);
        xc1 = wmma4(aG1, bR, xc1);
      }
    }
  }

  float* xp = xyz_ws + (size_t)atom * 100;
  if (r < 4) {
#pragma unroll
    for (int vr = 0; vr < 8; ++vr) {
      const int e = vr + 8 * g;
      xp[r * 25 + e] = SCALEF * xc0[vr];
      const int e2 = 16 + e;
      if (e2 < EMBC) xp[r * 25 + e2] = SCALEF * xc1[vr];
    }
  }
}

// ---------------------------------------------------------------- K2: DR = xyz . xyz[:, :16]
__global__ void k_dr(const float* __restrict__ xyz, float* __restrict__ dr) {
  const int idx = blockIdx.x * blockDim.x + threadIdx.x;
  if (idx >= BB * NAA * DDC) return;
  const int atom = idx / DDC;
  const int ef = idx % DDC;
  const int e = ef >> 4, f = ef & 15;
  const float* xp = xyz + (size_t)atom * 100;
  float s = 0.f;
#pragma unroll
  for (int c = 0; c < 4; c++) s += xp[c * 25 + e] * xp[c * 25 + f];
  dr[idx] = s;
}

// ---------------------------------------------------------------- K3: fitting net fwd + bwd
// one wave per 16 atoms of one type
__global__ void __launch_bounds__(128) k_fit(
    const float* __restrict__ drws,
    const float* __restrict__ fw0, const float* __restrict__ fb0,
    const float* __restrict__ fw1, const float* __restrict__ fb1,
    const float* __restrict__ fw2, const float* __restrict__ fb2,
    const float* __restrict__ fwo, const float* __restrict__ fbo,
    float* o_etot, float* __restrict__ o_ei, float* __restrict__ gdr)
{
  __shared__ float sT[4][4][16][52];
  const int wave = threadIdx.x >> 5, lane = threadIdx.x & 31;
  const int g = lane >> 4, r = lane & 15;
  const int t = blockIdx.x >> 5, blk = blockIdx.x & 31;
  const int wb = (blk * 4 + wave) * 16;
  const int i2 = wb + r;
  const int b = i2 >> 9;
  const int n = t * NPTC + (i2 & 511);
  const size_t aidx = (size_t)b * NAA + n;
  const float* w0 = fw0 + t * DDC * FHC;
  const float* b0 = fb0 + t * FHC;
  const float* w1 = fw1 + t * FHC * FHC;
  const float* b1 = fb1 + t * FHC;
  const float* w2 = fw2 + t * FHC * FHC;
  const float* b2 = fb2 + t * FHC;
  const float* wo = fwo + t * FHC;
  const float bo = fbo[t];
  float (*T0)[52] = sT[wave][0];
  float (*T1)[52] = sT[wave][1];
  float (*T2)[52] = sT[wave][2];
  float (*G )[52] = sT[wave][3];
  const float* drrow = drws + aidx * DDC;

  // layer 0: DR(400) @ w0
  for (int nt = 0; nt < 4; ++nt) {
    const int nc = nt * 16 + r; const bool nv = nc < FHC;
    v8f acc;
#pragma unroll
    for (int i = 0; i < 8; i++) acc[i] = nv ? b0[nc] : 0.f;
    for (int kk = 0; kk < 100; ++kk) {
      const int k = kk * 4 + 2 * g;
      v2f av; av.x = drrow[k]; av.y = drrow[k + 1];
      v2f bv; bv.x = nv ? w0[k * FHC + nc] : 0.f; bv.y = nv ? w0[(k + 1) * FHC + nc] : 0.f;
      acc = wmma4(av, bv, acc);
    }
#pragma unroll
    for (int vr = 0; vr < 8; ++vr) { const int row = vr + 8 * g; if (nv) T0[row][nc] = tanhf(acc[vr]); }
  }
  // layer 1
  for (int nt = 0; nt < 4; ++nt) {
    const int nc = nt * 16 + r; const bool nv = nc < FHC;
    v8f acc;
#pragma unroll
    for (int i = 0; i < 8; i++) acc[i] = nv ? b1[nc] : 0.f;
    for (int kk = 0; kk < 13; ++kk) {
      const int k = kk * 4 + 2 * g;
      v2f av; av.x = (k < FHC) ? T0[r][k] : 0.f; av.y = (k + 1 < FHC) ? T0[r][k + 1] : 0.f;
      v2f bv; bv.x = (k < FHC && nv) ? w1[k * FHC + nc] : 0.f;
              bv.y = (k + 1 < FHC && nv) ? w1[(k + 1) * FHC + nc] : 0.f;
      acc = wmma4(av, bv, acc);
    }
#pragma unroll
    for (int vr = 0; vr < 8; ++vr) { const int row = vr + 8 * g; if (nv) T1[row][nc] = tanhf(acc[vr]); }
  }
  // layer 2 (A = h1 = T1 + T0)
  for (int nt = 0; nt < 4; ++nt) {
    const int nc = nt * 16 + r; const bool nv = nc < FHC;
    v8f acc;
#pragma unroll
    for (int i = 0; i < 8; i++) acc[i] = nv ? b2[nc] : 0.f;
    for (int kk = 0; kk < 13; ++kk) {
      const int k = kk * 4 + 2 * g;
      v2f av; av.x = (k < FHC) ? (T1[r][k] + T0[r][k]) : 0.f;
              av.y = (k + 1 < FHC) ? (T1[r][k + 1] + T0[r][k + 1]) : 0.f;
      v2f bv; bv.x = (k < FHC && nv) ? w2[k * FHC + nc] : 0.f;
              bv.y = (k + 1 < FHC && nv) ? w2[(k + 1) * FHC + nc] : 0.f;
      acc = wmma4(av, bv, acc);
    }
#pragma unroll
    for (int vr = 0; vr < 8; ++vr) { const int row = vr + 8 * g; if (nv) T2[row][nc] = tanhf(acc[vr]); }
  }
  // Ei, Etot, gt2 (in place into T2)
  if (lane < 16) {
    float s = bo;
    for (int e = 0; e < FHC; ++e) s += (T0[r][e] + T1[r][e] + T2[r][e]) * wo[e];
    o_ei[aidx] = s;
    atomicAdd(&o_etot[b], s);
    for (int e = 0; e < FHC; ++e) { const float th = T2[r][e]; T2[r][e] = wo[e] * (1.f - th * th); }
  }
  // gh1 = gt2 @ w2^T + wo  -> G
  for (int nt = 0; nt < 4; ++nt) {
    const int nc = nt * 16 + r; const bool nv = nc < FHC;
    v8f acc;
#pragma unroll
    for (int i = 0; i < 8; i++) acc[i] = nv ? wo[nc] : 0.f;
    for (int kk = 0; kk < 13; ++kk) {
      const int k = kk * 4 + 2 * g;
      v2f av; av.x = (k < FHC) ? T2[r][k] : 0.f; av.y = (k + 1 < FHC) ? T2[r][k + 1] : 0.f;
      v2f bv; bv.x = (k < FHC && nv) ? w2[nc * FHC + k] : 0.f;
              bv.y = (k + 1 < FHC && nv) ? w2[nc * FHC + k + 1] : 0.f;
      acc = wmma4(av, bv, acc);
    }
#pragma unroll
    for (int vr = 0; vr < 8; ++vr) { const int row = vr + 8 * g; if (nv) G[row][nc] = acc[vr]; }
  }
  if (lane < 16)
    for (int e = 0; e < FHC; ++e) { const float th = T1[r][e]; T1[r][e] = G[r][e] * (1.f - th * th); }
  // gh0 = gt1 @ w1^T + gh1 -> G
  for (int nt = 0; nt < 4; ++nt) {
    const int nc = nt * 16 + r; const bool nv = nc < FHC;
    v8f acc;
#pragma unroll
    for (int i = 0; i < 8; i++) { const int row = i + 8 * g; acc[i] = nv ? G[row][nc] : 0.f; }
    for (int kk = 0; kk < 13; ++kk) {
      const int k = kk * 4 + 2 * g;
      v2f av; av.x = (k < FHC) ? T1[r][k] : 0.f; av.y = (k + 1 < FHC) ? T1[r][k + 1] : 0.f;
      v2f bv; bv.x = (k < FHC && nv) ? w1[nc * FHC + k] : 0.f;
              bv.y = (k + 1 < FHC && nv) ? w1[nc * FHC + k + 1] : 0.f;
      acc = wmma4(av, bv, acc);
    }
#pragma unroll
    for (int vr = 0; vr < 8; ++vr) { const int row = vr + 8 * g; if (nv) G[row][nc] = acc[vr]; }
  }
  if (lane < 16)
    for (int e = 0; e < FHC; ++e) { const float th = T0[r][e]; T0[r][e] = G[r][e] * (1.f - th * th); }
  // gDR = gt0 @ w0^T (16 x 400)
  for (int nt = 0; nt < 25; ++nt) {
    const int nc = nt * 16 + r;
    v8f acc;
#pragma unroll
    for (int i = 0; i < 8; i++) acc[i] = 0.f;
    for (int kk = 0; kk < 13; ++kk) {
      const int k = kk * 4 + 2 * g;
      v2f av; av.x = (k < FHC) ? T0[r][k] : 0.f; av.y = (k + 1 < FHC) ? T0[r][k + 1] : 0.f;
      v2f bv; bv.x = (k < FHC) ? w0[nc * FHC + k] : 0.f;
              bv.y = (k + 1 < FHC) ? w0[nc * FHC + k + 1] : 0.f;
      acc = wmma4(av, bv, acc);
    }
#pragma unroll
    for (int vr = 0; vr < 8; ++vr) {
      const int row = vr + 8 * g;
      const int i2r = wb + row;
      const int br = i2r >> 9;
      const int nr = t * NPTC + (i2r & 511);
      gdr[((size_t)br * NAA + nr) * DDC + nc] = acc[vr];
    }
  }
}

// ---------------------------------------------------------------- K4: gxyz from gDR
__global__ void k_gxyz(const float* __restrict__ xyz, const float* __restrict__ gdr,
                       float* __restrict__ gxyz) {
  const int idx = blockIdx.x * blockDim.x + threadIdx.x;
  if (idx >= BB * NAA * 100) return;
  const int atom = idx / 100;
  const int ce = idx % 100;
  const int c = ce / 25, e = ce % 25;
  const float* xp = xyz + (size_t)atom * 100 + c * 25;
  const float* gp = gdr + (size_t)atom * DDC;
  float s = 0.f;
#pragma unroll
  for (int f = 0; f < 16; ++f) s += gp[e * 16 + f] * xp[f];
  if (e < 16) {
#pragma unroll
    for (int e2 = 0; e2 < 25; ++e2) s += gp[e2 * 16 + e] * xp[e2];
  }
  gxyz[idx] = s;
}

// ---------------------------------------------------------------- K5: embedding bwd + forces + virial
__global__ void __launch_bounds__(128) k_emb_bwd(
    const float* __restrict__ Ri, const float* __restrict__ ImageDR,
    const float* __restrict__ dfeat, const int* __restrict__ list_neigh,
    const float* __restrict__ ew0, const float* __restrict__ eb0,
    const float* __restrict__ ew1, const float* __restrict__ eb1,
    const float* __restrict__ ew2, const float* __restrict__ eb2,
    const float* __restrict__ gxyz, float* o_F, float* o_vir)
{
  __shared__ float sH[4][5][16][26];   // H0, T1, T2, G2, GH
  __shared__ float sGX[4][4][26];
  __shared__ float sR[4][16][4];
  const int wave = threadIdx.x >> 5, lane = threadIdx.x & 31;
  const int g = lane >> 4, r = lane & 15;
  const int atom = blockIdx.x * 4 + wave;
  const int b = atom >> 10;
  const int n = atom & (NAA - 1);
  const int t = (n >= NPTC) ? 1 : 0;
  const int c1 = 16 + r;
  float (*H0)[26] = sH[wave][0];
  float (*T1)[26] = sH[wave][1];
  float (*T2)[26] = sH[wave][2];
  float (*G2)[26] = sH[wave][3];
  float (*GH)[26] = sH[wave][4];
  float (*GX)[26] = sGX[wave];
  float (*Rt)[4]  = sR[wave];

  for (int idx = lane; idx < 100; idx += 32) GX[idx / 25][idx % 25] = gxyz[(size_t)atom * 100 + idx];
  const float* ribase = Ri + (size_t)atom * MMC * 4;
  float vir[9];
#pragma unroll
  for (int k = 0; k < 9; k++) vir[k] = 0.f;

  for (int t1 = 0; t1 < 2; ++t1) {
    const int p = t * 2 + t1;
    const float* w0 = ew0 + p * EMBC;
    const float* b0 = eb0 + p * EMBC;
    const float* W1 = ew1 + p * EMBC * EMBC;
    const float* B1 = eb1 + p * EMBC;
    const float* W2 = ew2 + p * EMBC * EMBC;
    const float* B2 = eb2 + p * EMBC;

    for (int tile = 0; tile < 7; ++tile) {
      const int mrow = tile * 16 + r;
      const bool valid = (mrow < MAXNC);
      const int mglob = t1 * MAXNC + mrow;
      if (lane < 16) {
        float4 rv = valid ? *(const float4*)(ribase + (size_t)mglob * 4)
                          : make_float4(0.f, 0.f, 0.f, 0.f);
        Rt[r][0] = rv.x; Rt[r][1] = rv.y; Rt[r][2] = rv.z; Rt[r][3] = rv.w;
      }
      const float S = valid ? ribase[(size_t)mglob * 4] : 0.f;
      float h0[EMBC];
#pragma unroll
      for (int e = 0; e < EMBC; e++) h0[e] = tanhf(S * w0[e] + b0[e]);
      if (lane < 16) {
#pragma unroll
        for (int e = 0; e < EMBC; e++) H0[r][e] = h0[e];
      }
      // layer1 fwd -> tanh only into T1
      {
        v8f a0, a1;
#pragma unroll
        for (int i = 0; i < 8; i++) { a0[i] = B1[r]; a1[i] = (c1 < EMBC) ? B1[c1] : 0.f; }
        for (int kk = 0; kk < 7; ++kk) {
          const int k = kk * 4 + 2 * g;
          v2f av;  av.x = (k < EMBC) ? h0[k] : 0.f; av.y = (k + 1 < EMBC) ? h0[k + 1] : 0.f;
          v2f bv0; bv0.x = (k < EMBC) ? W1[k * EMBC + r] : 0.f;
                   bv0.y = (k + 1 < EMBC) ? W1[(k + 1) * EMBC + r] : 0.f;
          v2f bv1; bv1.x = (k < EMBC && c1 < EMBC) ? W1[k * EMBC + c1] : 0.f;
                   bv1.y = (k + 1 < EMBC && c1 < EMBC) ? W1[(k + 1) * EMBC + c1] : 0.f;
          a0 = wmma4(av, bv0, a0); a1 = wmma4(av, bv1, a1);
        }
#pragma unroll
        for (int vr = 0; vr < 8; ++vr) {
          const int row = vr + 8 * g;
          T1[row][r] = tanhf(a0[vr]);
          if (c1 < EMBC) T1[row][c1] = tanhf(a1[vr]);
        }
      }
      float h1v[EMBC];
#pragma unroll
      for (int e = 0; e < EMBC; e++) h1v[e] = T1[r][e] + h0[e];
      // layer2 fwd -> tanh only into T2
      {
        v8f a0, a1;
#pragma unroll
        for (int i = 0; i < 8; i++) { a0[i] = B2[r]; a1[i] = (c1 < EMBC) ? B2[c1] : 0.f; }
        for (int kk = 0; kk < 7; ++kk) {
          const int k = kk * 4 + 2 * g;
          v2f av;  av.x = (k < EMBC) ? h1v[k] : 0.f; av.y = (k + 1 < EMBC) ? h1v[k + 1] : 0.f;
          v2f bv0; bv0.x = (k < EMBC) ? W2[k * EMBC + r] : 0.f;
                   bv0.y = (k + 1 < EMBC) ? W2[(k + 1) * EMBC + r] : 0.f;
          v2f bv1; bv1.x = (k < EMBC && c1 < EMBC) ? W2[k * EMBC + c1] : 0.f;
                   bv1.y = (k + 1 < EMBC && c1 < EMBC) ? W2[(k + 1) * EMBC + c1] : 0.f;
          a0 = wmma4(av, bv0, a0); a1 = wmma4(av, bv1, a1);
        }
#pragma unroll
        for (int vr = 0; vr < 8; ++vr) {
          const int row = vr + 8 * g;
          T2[row][r] = tanhf(a0[vr]);
          if (c1 < EMBC) T2[row][c1] = tanhf(a1[vr]);
        }
      }
      float Gr[EMBC], g2[EMBC], gt2[EMBC];
#pragma unroll
      for (int e = 0; e < EMBC; e++) {
        const float th2 = T2[r][e];
        Gr[e] = th2 + h1v[e];
        g2[e] = SCALEF * (Rt[r][0] * GX[0][e] + Rt[r][1] * GX[1][e] +
                          Rt[r][2] * GX[2][e] + Rt[r][3] * GX[3][e]);
        gt2[e] = g2[e] * (1.f - th2 * th2);
      }
      if (lane < 16) {
#pragma unroll
        for (int e = 0; e < EMBC; e++) G2[r][e] = g2[e];
      }
      // gh1 = gt2 @ W2^T + g2 -> GH
      {
        v8f a0, a1;
#pragma unroll
        for (int i = 0; i < 8; i++) {
          const int row = i + 8 * g;
          a0[i] = G2[row][r]; a1[i] = (c1 < EMBC) ? G2[row][c1] : 0.f;
        }
        for (int kk = 0; kk < 7; ++kk) {
          const int k = kk * 4 + 2 * g;
          v2f av;  av.x = (k < EMBC) ? gt2[k] : 0.f; av.y = (k + 1 < EMBC) ? gt2[k + 1] : 0.f;
          v2f bv0; bv0.x = (k < EMBC) ? W2[r * EMBC + k] : 0.f;
                   bv0.y = (k + 1 < EMBC) ? W2[r * EMBC + k + 1] : 0.f;
          v2f bv1; bv1.x = (k < EMBC && c1 < EMBC) ? W2[c1 * EMBC + k] : 0.f;
                   bv1.y = (k + 1 < EMBC && c1 < EMBC) ? W2[c1 * EMBC + k + 1] : 0.f;
          a0 = wmma4(av, bv0, a0); a1 = wmma4(av, bv1, a1);
        }
#pragma unroll
        for (int vr = 0; vr < 8; ++vr) {
          const int row = vr + 8 * g;
          GH[row][r] = a0[vr];
          if (c1 < EMBC) GH[row][c1] = a1[vr];
        }
      }
      float gt1[EMBC];
#pragma unroll
      for (int e = 0; e < EMBC; e++) { const float th = T1[r][e]; gt1[e] = GH[r][e] * (1.f - th * th); }
      // gh0 = gt1 @ W1^T + gh1 -> GH
      {
        v8f a0, a1;
#pragma unroll
        for (int i = 0; i < 8; i++) {
          const int row = i + 8 * g;
          a0[i] = GH[row][r]; a1[i] = (c1 < EMBC) ? GH[row][c1] : 0.f;
        }
        for (int kk = 0; kk < 7; ++kk) {
          const int k = kk * 4 + 2 * g;
          v2f av;  av.x = (k < EMBC) ? gt1[k] : 0.f; av.y = (k + 1 < EMBC) ? gt1[k + 1] : 0.f;
          v2f bv0; bv0.x = (k < EMBC) ? W1[r * EMBC + k] : 0.f;
                   bv0.y = (k + 1 < EMBC) ? W1[r * EMBC + k + 1] : 0.f;
          v2f bv1; bv1.x = (k < EMBC && c1 < EMBC) ? W1[c1 * EMBC + k] : 0.f;
                   bv1.y = (k + 1 < EMBC && c1 < EMBC) ? W1[c1 * EMBC + k + 1] : 0.f;
          a0 = wmma4(av, bv0, a0); a1 = wmma4(av, bv1, a1);
        }
#pragma unroll
        for (int vr = 0; vr < 8; ++vr) {
          const int row = vr + 8 * g;
          GH[row][r] = a0[vr];
          if (c1 < EMBC) GH[row][c1] = a1[vr];
        }
      }
      // per-neighbor dE, forces, virial
      if (lane < 16 && valid) {
        float gS = 0.f;
#pragma unroll
        for (int e = 0; e < EMBC; e++) gS += GH[r][e] * (1.f - h0[e] * h0[e]) * w0[e];
        float dE[4];
#pragma unroll
        for (int c = 0; c < 4; ++c) {
          float s = 0.f;
#pragma unroll
          for (int e = 0; e < EMBC; e++) s += GX[c][e] * Gr[e];
          dE[c] = SCALEF * s;
        }
        dE[0] += gS;
        const float* df = dfeat + ((size_t)atom * 800 + (size_t)mglob * 4) * 3;
        const int j = list_neigh[(size_t)atom * MMC + mglob] - 1;
        const float* im = ImageDR + ((size_t)atom * MMC + mglob) * 3;
#pragma unroll
        for (int x = 0; x < 3; ++x) {
          const float fx = dE[0] * df[x] + dE[1] * df[3 + x] + dE[2] * df[6 + x] + dE[3] * df[9 + x];
          atomicAdd(&o_F[(size_t)atom * 3 + x], -fx);
          if (j >= 0) {
            atomicAdd(&o_F[((size_t)b * NAA + j) * 3 + x], fx);
#pragma unroll
            for (int a = 0; a < 3; ++a) vir[a * 3 + x] -= im[a] * fx;
          }
        }
      }
    }
  }
  if (lane < 16) {
#pragma unroll
    for (int k = 0; k < 9; ++k) atomicAdd(&o_vir[b * 9 + k], vir[k]);
  }
}

// ---------------------------------------------------------------- K6: Egroup
__global__ void __launch_bounds__(128) k_egroup(
    const float* __restrict__ W, const float* __restrict__ divd,
    const float* __restrict__ Ei, float* __restrict__ outEg)
{
  const int b = blockIdx.x >> 10;
  const int i = blockIdx.x & (NAA - 1);
  __shared__ float red[128];
  const float* row = W + ((size_t)b * NAA + i) * NAA;
  const float* ei = Ei + (size_t)b * NAA;
  float s = 0.f;
  for (int j = threadIdx.x; j < NAA; j += 128) s += row[j] * ei[j];
  red[threadIdx.x] = s;
  __syncthreads();
  for (int off = 64; off > 0; off >>= 1) {
    if (threadIdx.x < off) red[threadIdx.x] += red[threadIdx.x + off];
    __syncthreads();
  }
  if (threadIdx.x == 0) outEg[(size_t)b * NAA + i] = red[0] / divd[(size_t)b * NAA + i];
}

// ---------------------------------------------------------------- launcher
extern "C" void kernel_launch(void* const* d_in, const int* in_sizes, int n_in,
                              void* d_out, int out_size, void* d_ws, size_t ws_size,
                              hipStream_t stream) {
  const float* ImageDR    = (const float*)d_in[0];
  const float* Ri         = (const float*)d_in[1];
  const float* dfeat      = (const float*)d_in[2];
  const int*   list_neigh = (const int*)  d_in[3];
  // d_in[4] natoms_img: fixed [NA, 512, 512]
  const float* Egw     = (const float*)d_in[5];
  const float* divider = (const float*)d_in[6];
  const float* ew0 = (const float*)d_in[7];
  const float* eb0 = (const float*)d_in[8];
  const float* ew1 = (const float*)d_in[9];
  const float* eb1 = (const float*)d_in[10];
  const float* ew2 = (const float*)d_in[11];
  const float* eb2 = (const float*)d_in[12];
  const float* fw0 = (const float*)d_in[13];
  const float* fb0 = (const float*)d_in[14];
  const float* fw1 = (const float*)d_in[15];
  const float* fb1 = (const float*)d_in[16];
  const float* fw2 = (const float*)d_in[17];
  const float* fb2 = (const float*)d_in[18];
  const float* fwo = (const float*)d_in[19];
  const float* fbo = (const float*)d_in[20];

  float* out = (float*)d_out;
  float* ws  = (float*)d_ws;
  float* xyz  = ws;                 // 409600
  float* gxyz = ws + 409600;        // 409600
  float* dr   = ws + 819200;        // 1638400
  float* gdr  = ws + 2457600;       // 1638400

  float* o_etot = out;              // 4
  float* o_ei   = out + 4;          // 4096
  float* o_F    = out + 4100;       // 12288
  float* o_eg   = out + 16388;      // 4096
  float* o_vir  = out + 20484;      // 36
  const int out_n = 20520;

  k_zero<<<(out_n + 255) / 256, 256, 0, stream>>>(out, out_n);
  k_emb_fwd<<<1024, 128, 0, stream>>>(Ri, ew0, eb0, ew1, eb1, ew2, eb2, xyz);
  k_dr<<<(BB * NAA * DDC + 255) / 256, 256, 0, stream>>>(xyz, dr);
  k_fit<<<64, 128, 0, stream>>>(dr, fw0, fb0, fw1, fb1, fw2, fb2, fwo, fbo, o_etot, o_ei, gdr);
  k_gxyz<<<(BB * NAA * 100 + 255) / 256, 256, 0, stream>>>(xyz, gdr, gxyz);
  k_emb_bwd<<<1024, 128, 0, stream>>>(Ri, ImageDR, dfeat, list_neigh,
                                      ew0, eb0, ew1, eb1, ew2, eb2, gxyz, o_F, o_vir);
  k_egroup<<<BB * NAA, 128, 0, stream>>>(Egw, divider, o_ei, o_eg);
}